// LegacyFullEwaldMeshLongRangeKernel3D_70686571757834
// MI455X (gfx1250) — compile-verified
//
#include <hip/hip_runtime.h>
#include <math.h>

// MI455X / gfx1250 implementation of the full-Ewald mesh long-range kernel.
//  - spread_kernel : CIC scatter into a 128KB LDS mesh (ds_add_f32), + source sum
//  - fft_kernel    : 6 in-LDS 32-point DFT passes (256KB complex mesh in LDS),
//                    spectral weights computed analytically in between
//  - pair_kernel   : real-space erfc sum; 16x16 tile cross-terms via
//                    v_wmma_f32_16x16x4_f32; shift corrections are rank-1 and
//                    come from an LDS-resident 27-entry shift table; erfc is a
//                    branch-free A&S 7.1.26 polynomial (arg always >= 0).
#define MM    32
#define MESH  (MM*MM*MM)
#define NB    2
#define NA    1024
#define NT    (NA/16)
#define PI_F  3.14159265358979323846f
#define EPS_F 1e-12f
#define FLOOR_F 1e-6f

typedef __attribute__((ext_vector_type(2))) float v2f;
typedef __attribute__((ext_vector_type(8))) float v8f;

__device__ __forceinline__ void inv3x3(const float* c, float* inv, float* det_out) {
  float a00=c[0],a01=c[1],a02=c[2],a10=c[3],a11=c[4],a12=c[5],a20=c[6],a21=c[7],a22=c[8];
  float det = a00*(a11*a22-a12*a21) - a01*(a10*a22-a12*a20) + a02*(a10*a21-a11*a20);
  float id = 1.0f/det;
  inv[0]=(a11*a22-a12*a21)*id; inv[1]=(a02*a21-a01*a22)*id; inv[2]=(a01*a12-a02*a11)*id;
  inv[3]=(a12*a20-a10*a22)*id; inv[4]=(a00*a22-a02*a20)*id; inv[5]=(a02*a10-a00*a12)*id;
  inv[6]=(a10*a21-a11*a20)*id; inv[7]=(a01*a20-a00*a21)*id; inv[8]=(a00*a11-a01*a10)*id;
  *det_out = det;
}

__device__ __forceinline__ float graph_cutoff(const float* c) {
  float r0 = sqrtf(c[0]*c[0]+c[1]*c[1]+c[2]*c[2]);
  float r1 = sqrtf(c[3]*c[3]+c[4]*c[4]+c[5]*c[5]);
  float r2 = sqrtf(c[6]*c[6]+c[7]*c[7]+c[8]*c[8]);
  return fmaxf(0.5f*fminf(r0,fminf(r1,r2)), FLOOR_F);
}

// Branch-free erfc for x >= 0 (Abramowitz & Stegun 7.1.26, |err| < 1.5e-7).
__device__ __forceinline__ float erfc_pos(float x) {
  const float t = 1.0f/(1.0f + 0.3275911f*x);
  float p =  1.061405429f;
  p = p*t - 1.453152027f;
  p = p*t + 1.421413741f;
  p = p*t - 0.284496736f;
  p = p*t + 0.254829592f;
  return p*t*__expf(-x*x);
}

// ---------------------------------------------------------------------------
// Kernel 1: CIC spread into LDS mesh, one workgroup per graph (1024 threads).
// ---------------------------------------------------------------------------
__global__ __launch_bounds__(1024)
void spread_kernel(const float* __restrict__ pos, const float* __restrict__ cell,
                   const float* __restrict__ src, float* __restrict__ meshR,
                   float* __restrict__ srcsum) {
  extern __shared__ float meshL[];               // MESH floats (128 KB)
  __shared__ float sred[32];
  const int g = blockIdx.x;
  const int t = threadIdx.x;
  for (int i = t; i < MESH; i += 1024) meshL[i] = 0.0f;
  __syncthreads();

  float inv[9], det;
  inv3x3(cell + g*9, inv, &det);

  const int ai = g*NA + t;
  const float p0 = pos[ai*3+0], p1 = pos[ai*3+1], p2 = pos[ai*3+2];
  const float s  = src[ai];

  float fr[3];
  #pragma unroll
  for (int j = 0; j < 3; ++j) {
    float f = p0*inv[j] + p1*inv[3+j] + p2*inv[6+j];
    f -= floorf(f);                               // mod 1.0
    fr[j] = f * (float)MM;
  }
  const int bx=(int)floorf(fr[0]), by=(int)floorf(fr[1]), bz=(int)floorf(fr[2]);
  const float fx=fr[0]-(float)bx, fy=fr[1]-(float)by, fz=fr[2]-(float)bz;
  #pragma unroll
  for (int o = 0; o < 8; ++o) {
    const int ox=(o>>2)&1, oy=(o>>1)&1, oz=o&1;
    const int ix=(bx+ox)&(MM-1), iy=(by+oy)&(MM-1), iz=(bz+oz)&(MM-1);
    const float w = (ox?fx:1.0f-fx)*(oy?fy:1.0f-fy)*(oz?fz:1.0f-fz);
    atomicAdd(&meshL[(ix*MM+iy)*MM+iz], w*s);
  }

  float v = s;                                    // per-graph source sum
  #pragma unroll
  for (int m = 16; m >= 1; m >>= 1) v += __shfl_xor(v, m, 32);
  if ((t & 31) == 0) sred[t >> 5] = v;
  __syncthreads();                                // also orders the LDS atomics
  if (t == 0) { float tot=0.0f; for (int k=0;k<32;++k) tot+=sred[k]; srcsum[g]=tot; }
  for (int i = t; i < MESH; i += 1024) meshR[g*MESH + i] = meshL[i];
}

// ---------------------------------------------------------------------------
// Kernel 2: fused FFT * sw * IFFT, complex mesh resident in 256 KB of LDS.
// ---------------------------------------------------------------------------
__device__ __forceinline__ void dft_pass(float* re, float* im,
                                         const float* twc, const float* tws,
                                         int axis, float sgn, int t) {
  for (int l = t; l < MM*MM; l += 256) {
    int base, stride;
    if (axis == 2)      { base = l*MM;                      stride = 1;     }
    else if (axis == 1) { base = (l>>5)*MM*MM + (l&31);     stride = MM;    }
    else                { base = (l>>5)*MM + (l&31);        stride = MM*MM; }
    float ir[MM], ii[MM];
    #pragma unroll
    for (int z = 0; z < MM; ++z) { ir[z]=re[base+z*stride]; ii[z]=im[base+z*stride]; }
    for (int k = 0; k < MM; ++k) {
      float sr = 0.0f, si = 0.0f;
      #pragma unroll
      for (int z = 0; z < MM; ++z) {
        const int tt = (k*z) & (MM-1);
        const float c = twc[tt], sn = tws[tt];
        sr += ir[z]*c + sgn*ii[z]*sn;             // sgn=+1: e^{-i}, sgn=-1: e^{+i}
        si += ii[z]*c - sgn*ir[z]*sn;
      }
      re[base+k*stride] = sr; im[base+k*stride] = si;
    }
  }
}

__global__ __launch_bounds__(256)
void fft_kernel(const float* __restrict__ cell, float* __restrict__ meshR) {
  extern __shared__ float sm[];                   // re[MESH] ++ im[MESH]
  float* re = sm;
  float* im = sm + MESH;
  __shared__ float twc[MM], tws[MM], sinc2t[MM];
  const int g = blockIdx.x, t = threadIdx.x;

  if (t < MM) {
    const float ang = 2.0f*PI_F*(float)t/(float)MM;
    twc[t] = cosf(ang); tws[t] = sinf(ang);
    const float f = (t < MM/2) ? (float)t : (float)(t - MM);
    const float x = f/(float)MM;
    float sc = 1.0f;
    if (x != 0.0f) { const float px = PI_F*x; sc = sinf(px)/px; }
    sinc2t[t] = sc*sc;
  }
  for (int i = t; i < MESH; i += 256) { re[i] = meshR[g*MESH+i]; im[i] = 0.0f; }
  __syncthreads();

  dft_pass(re, im, twc, tws, 2, 1.0f, t); __syncthreads();
  dft_pass(re, im, twc, tws, 1, 1.0f, t); __syncthreads();
  dft_pass(re, im, twc, tws, 0, 1.0f, t); __syncthreads();

  // spectral weights
  float inv[9], det; inv3x3(cell + g*9, inv, &det);
  const float vol    = fmaxf(fabsf(det), FLOOR_F);
  const float cutoff = graph_cutoff(cell + g*9);
  const float alpha  = 5.0f / fmaxf(cutoff, FLOOR_F);
  const float inv4a2 = 1.0f/(4.0f*alpha*alpha);
  for (int i = t; i < MESH; i += 256) {
    const int a = i >> 10, b = (i >> 5) & 31, c = i & 31;
    const float fa = (a<16)?(float)a:(float)(a-32);
    const float fb = (b<16)?(float)b:(float)(b-32);
    const float fc = (c<16)?(float)c:(float)(c-32);
    const float kx = 2.0f*PI_F*(fa*inv[0]+fb*inv[3]+fc*inv[6]);
    const float ky = 2.0f*PI_F*(fa*inv[1]+fb*inv[4]+fc*inv[7]);
    const float kz = 2.0f*PI_F*(fa*inv[2]+fb*inv[5]+fc*inv[8]);
    const float ksq = kx*kx+ky*ky+kz*kz;
    const float kn  = sqrtf(fmaxf(ksq, EPS_F));
    const float kd  = fmaxf(kn, FLOOR_F);
    float sw = (4.0f*PI_F/(kd*kd)) / vol * expf(-kn*kn*inv4a2);
    const float wd = fmaxf(sinc2t[a]*sinc2t[b]*sinc2t[c], FLOOR_F);
    sw = sw/(wd*wd);
    sw = (kn > FLOOR_F) ? sw : 0.0f;
    sw *= (1.0f/(float)MESH);                     // ifftn normalization folded in
    re[i] *= sw; im[i] *= sw;
  }
  __syncthreads();

  dft_pass(re, im, twc, tws, 2, -1.0f, t); __syncthreads();
  dft_pass(re, im, twc, tws, 1, -1.0f, t); __syncthreads();
  dft_pass(re, im, twc, tws, 0, -1.0f, t); __syncthreads();

  for (int i = t; i < MESH; i += 256) meshR[g*MESH+i] = re[i];   // pot_mesh (real)
}

// ---------------------------------------------------------------------------
// Kernel 3: real-space erfc pair sum (WMMA tile cross-terms) + finalize.
// One block per (graph, 16-row i-tile); 8 waves x 8 j-tiles each.
// ---------------------------------------------------------------------------
__global__ __launch_bounds__(256)
void pair_kernel(const float* __restrict__ pos, const float* __restrict__ cell,
                 const float* __restrict__ src, const float* __restrict__ meshR,
                 const float* __restrict__ srcsum, float* __restrict__ out) {
  __shared__ float posL[NA*3];
  __shared__ float srcL[NA];
  __shared__ float rowpart[8][16];
  __shared__ float shtab[27][4];                  // cartesian shifts + |s|^2
  const int g  = blockIdx.x / NT;
  const int ti = blockIdx.x % NT;
  const int t  = threadIdx.x;

  __builtin_prefetch(meshR + g*MESH, 0, 1);       // global_prefetch_b8 warm-up
  for (int i = t; i < NA*3; i += 256) posL[i] = pos[g*NA*3 + i];
  for (int i = t; i < NA;   i += 256) srcL[i] = src[g*NA + i];

  float cr[9];
  #pragma unroll
  for (int i = 0; i < 9; ++i) cr[i] = cell[g*9 + i];
  const float cutoff = graph_cutoff(cr);
  const float alpha  = 5.0f / fmaxf(cutoff, FLOOR_F);

  if (t < 27) {                                   // block-invariant shift table
    const float shx = (float)(t/9 - 1);
    const float shy = (float)((t/3)%3 - 1);
    const float shz = (float)(t%3 - 1);
    const float sx = shx*cr[0] + shy*cr[3] + shz*cr[6];
    const float sy = shx*cr[1] + shy*cr[4] + shz*cr[7];
    const float sz = shx*cr[2] + shy*cr[5] + shz*cr[8];
    shtab[t][0] = sx; shtab[t][1] = sy; shtab[t][2] = sz;
    shtab[t][3] = sx*sx + sy*sy + sz*sz;
  }
  __syncthreads();

  const int w    = t >> 5;                        // wave in block
  const int l    = t & 31;                        // lane
  const int half = l >> 4;
  const int nn   = l & 15;                        // N index of this lane
  const int i0   = ti*16;

  // A-tile rows this lane owns in the C/D layout: M = v + 8*half
  float amx[8], amy[8], amz[8], am2[8];
  #pragma unroll
  for (int v = 0; v < 8; ++v) {
    const int m = v + 8*half;
    amx[v]=posL[(i0+m)*3+0]; amy[v]=posL[(i0+m)*3+1]; amz[v]=posL[(i0+m)*3+2];
    am2[v]=amx[v]*amx[v]+amy[v]*amy[v]+amz[v]*amz[v];
  }
  // WMMA A operand (16x4 f32): v0 = K0 (lanes 0-15) / K2 (16-31), v1 = K1 / K3(=0)
  const float ax = posL[(i0+nn)*3+0], ay = posL[(i0+nn)*3+1], az = posL[(i0+nn)*3+2];
  v2f A; A.x = half ? az : ax; A.y = half ? 0.0f : ay;

  float wacc[8];
  #pragma unroll
  for (int v = 0; v < 8; ++v) wacc[v] = 0.0f;

  for (int q = 0; q < 8; ++q) {
    const int jt = w + 8*q;
    const int j0 = jt*16;
    const float bx = posL[(j0+nn)*3+0], by = posL[(j0+nn)*3+1], bz = posL[(j0+nn)*3+2];
    const float bn2  = bx*bx+by*by+bz*bz;
    const float srcN = srcL[j0+nn];
    v2f Bv; Bv.x = half ? bz : bx; Bv.y = half ? 0.0f : by;

    v8f c0 = {0.0f,0.0f,0.0f,0.0f,0.0f,0.0f,0.0f,0.0f};
    // dotAB[v] = pos_i[i0 + v + 8*half] . pos_j[j0 + nn]
    v8f dotAB = __builtin_amdgcn_wmma_f32_16x16x4_f32(
        false, A, false, Bv, (short)0, c0, false, false);

    float acc[8];
    #pragma unroll
    for (int v = 0; v < 8; ++v) acc[v] = 0.0f;

    for (int sidx = 0; sidx < 27; ++sidx) {
      const float sx = shtab[sidx][0];
      const float sy = shtab[sidx][1];
      const float sz = shtab[sidx][2];
      const float s2 = shtab[sidx][3];
      const float cN = bn2 + 2.0f*(bx*sx + by*sy + bz*sz) + s2;
      const bool zshift = (sidx == 13);
      #pragma unroll
      for (int v = 0; v < 8; ++v) {
        const float d2 = am2[v] - 2.0f*(amx[v]*sx + amy[v]*sy + amz[v]*sz)
                       + cN - 2.0f*dotAB[v];
        const float r    = sqrtf(fmaxf(d2, EPS_F));
        const float kern = erfc_pos(alpha*r) / fmaxf(r, FLOOR_F);
        const int   m    = v + 8*half;
        const bool  self  = zshift && ((i0+m) == (j0+nn));
        const bool  valid = (r <= cutoff) && !self;
        acc[v] += valid ? kern : 0.0f;
      }
    }
    #pragma unroll
    for (int v = 0; v < 8; ++v) wacc[v] += acc[v]*srcN;   // fold source[j]
  }

  // reduce over N (16-lane groups), deterministic tree
  #pragma unroll
  for (int v = 0; v < 8; ++v) {
    float x = wacc[v];
    x += __shfl_xor(x, 1, 32);
    x += __shfl_xor(x, 2, 32);
    x += __shfl_xor(x, 4, 32);
    x += __shfl_xor(x, 8, 32);
    if (nn == 0) rowpart[w][v + 8*half] = x;
  }
  __syncthreads();

  if (t < 16) {
    float realp = 0.0f;
    #pragma unroll
    for (int ww = 0; ww < 8; ++ww) realp += rowpart[ww][t];   // fixed order

    float inv[9], det; inv3x3(cr, inv, &det);
    const float vol = fmaxf(fabsf(det), FLOOR_F);
    const int   ia  = i0 + t;
    const float p0 = posL[ia*3+0], p1 = posL[ia*3+1], p2 = posL[ia*3+2];
    float fr[3];
    #pragma unroll
    for (int j = 0; j < 3; ++j) {
      float f = p0*inv[j] + p1*inv[3+j] + p2*inv[6+j];
      f -= floorf(f);
      fr[j] = f*(float)MM;
    }
    const int bx=(int)floorf(fr[0]), by=(int)floorf(fr[1]), bz=(int)floorf(fr[2]);
    const float fx=fr[0]-(float)bx, fy=fr[1]-(float)by, fz=fr[2]-(float)bz;
    float pot = 0.0f;
    #pragma unroll
    for (int o = 0; o < 8; ++o) {
      const int ox=(o>>2)&1, oy=(o>>1)&1, oz=o&1;
      const int ix=(bx+ox)&(MM-1), iy=(by+oy)&(MM-1), iz=(bz+oz)&(MM-1);
      const float wg = (ox?fx:1.0f-fx)*(oy?fy:1.0f-fy)*(oz?fz:1.0f-fz);
      pot += meshR[g*MESH + (ix*MM+iy)*MM+iz]*wg;
    }
    const float si   = srcL[ia];
    const float self = -2.0f*alpha/sqrtf(PI_F)*si;
    const float bg   = -PI_F*srcsum[g]/(alpha*alpha*vol);
    out[g*NA + ia] = 0.5f*si*(pot + realp + self + bg);
  }
}

// ---------------------------------------------------------------------------
extern "C" void kernel_launch(void* const* d_in, const int* in_sizes, int n_in,
                              void* d_out, int out_size, void* d_ws, size_t ws_size,
                              hipStream_t stream) {
  (void)in_sizes; (void)n_in; (void)out_size; (void)ws_size;
  const float* pos  = (const float*)d_in[0];
  // d_in[1] = batch (int64) : graphs are equal-size & sorted, not needed
  const float* cell = (const float*)d_in[2];
  const float* src  = (const float*)d_in[3];
  float* out    = (float*)d_out;
  float* meshR  = (float*)d_ws;                   // [NB][MESH]
  float* srcsum = meshR + NB*MESH;                // [NB]

  hipFuncSetAttribute((const void*)spread_kernel,
                      hipFuncAttributeMaxDynamicSharedMemorySize, MESH*(int)sizeof(float));
  hipFuncSetAttribute((const void*)fft_kernel,
                      hipFuncAttributeMaxDynamicSharedMemorySize, 2*MESH*(int)sizeof(float));

  spread_kernel<<<NB, 1024, MESH*sizeof(float), stream>>>(pos, cell, src, meshR, srcsum);
  fft_kernel<<<NB, 256, 2*MESH*sizeof(float), stream>>>(cell, meshR);
  pair_kernel<<<NB*NT, 256, 0, stream>>>(pos, cell, src, meshR, srcsum, out);
}